// LinearSystemGroup_76828374991483
// MI455X (gfx1250) — compile-verified
//
#include <hip/hip_runtime.h>

// ---------------------------------------------------------------------------
// Chunked parallel scan for a batched linear state-space model on gfx1250.
//   G=2, BATCH=128, L=2048, S_D=64, I_D=16, O_D=16, all f32.
// All matmuls use V_WMMA_F32_16X16X4_F32 (full f32 precision, matches ref).
// Pipeline:
//   K0  powers:   F^1..F^64 per group via log-doubling       (ws, 2 MB)
//   K1  precomp:  U_t = u.B^T + w.S_W^T -> states region; V_t -> obs region
//   K2  chunk:    per 64-step chunk, partial scan with zero init (512 waves)
//   K3  carry:    sequential carries across 32 chunks using F^64 (16 waves)
//   K4  fixup:    s_t = carry.(F^T)^{j+1} + p_t ; o_t = s_t.H^T + V_t
// Workspace: 2*64*64*64 floats (powers) + 32*2*128*64 floats (carries) = 4 MB.
// ---------------------------------------------------------------------------

#define G_  2
#define B_  128
#define L_  2048
#define SD  64
#define ID  16
#define OD  16
#define LC  64          // chunk length
#define NC  (L_ / LC)   // 32 chunks
#define LDSLD 68        // padded LDS row stride (floats) -> conflict-free

typedef __attribute__((ext_vector_type(2))) float v2f;
typedef __attribute__((ext_vector_type(8))) float v8f;

static __device__ __forceinline__ v8f wmma4(v2f a, v2f b, v8f c) {
  // D = A(16x4) * B(4x16) + C(16x16), all f32, wave32
  return __builtin_amdgcn_wmma_f32_16x16x4_f32(false, a, false, b, (short)0, c,
                                               false, false);
}

static __device__ __forceinline__ v8f vzero8() {
  v8f v;
#pragma unroll
  for (int i = 0; i < 8; ++i) v[i] = 0.0f;
  return v;
}

// A-operand fragment (16x4, rows rowbase..rowbase+15, K-tile kt) from a
// row-major matrix with leading dim ld. Per ISA layout: lanes 0-15 hold
// K={4kt,4kt+1}, lanes 16-31 hold K={4kt+2,4kt+3}, lane%16 = row.
// The SAME pattern yields the B operand of x*M^T (B[k][n] = M[n][k]):
// lane%16 = n (column of result), K pair read along M's row.
static __device__ __forceinline__ v2f frag_T(const float* M, int ld,
                                             int rowbase, int kt) {
  const int lane = threadIdx.x & 31;
  const int l16  = lane & 15;
  const int koff = (lane >> 4) * 2;
  const float* p = M + (rowbase + l16) * ld + kt * 4 + koff;
  v2f r;
  r.x = p[0];
  r.y = p[1];
  return r;
}

// Non-transposed B operand: B[k][n] = M[k][n] (used only for F^a * F^b).
static __device__ __forceinline__ v2f frag_NT(const float* M, int ld,
                                              int colbase, int kt) {
  const int lane = threadIdx.x & 31;
  const int l16  = lane & 15;
  const int koff = (lane >> 4) * 2;
  const float* p = M + (kt * 4 + koff) * ld + colbase + l16;
  v2f r;
  r.x = p[0];
  r.y = p[ld];
  return r;
}

// C/D tile layout: VGPR r holds (row = r + 8*(lane/16), col = colbase+lane%16)
static __device__ __forceinline__ v8f ctile_load(const float* p, int ld,
                                                 int colbase) {
  const int lane  = threadIdx.x & 31;
  const int l16   = lane & 15;
  const int rbase = (lane >> 4) * 8;
  v8f c;
#pragma unroll
  for (int r = 0; r < 8; ++r) c[r] = p[(rbase + r) * ld + colbase + l16];
  return c;
}

static __device__ __forceinline__ void ctile_store(float* p, int ld,
                                                   int colbase, v8f c) {
  const int lane  = threadIdx.x & 31;
  const int l16   = lane & 15;
  const int rbase = (lane >> 4) * 8;
#pragma unroll
  for (int r = 0; r < 8; ++r) p[(rbase + r) * ld + colbase + l16] = c[r];
}

// Stage a 16x64 C/D-layout tile into padded LDS so it can be re-read as an
// A operand next step (D->A layout conversion = in-wave transpose via LDS).
static __device__ __forceinline__ void stage4(float* sm, const v8f* acc) {
  const int lane  = threadIdx.x & 31;
  const int l16   = lane & 15;
  const int rbase = (lane >> 4) * 8;
#pragma unroll
  for (int nt = 0; nt < 4; ++nt)
#pragma unroll
    for (int r = 0; r < 8; ++r)
      sm[(rbase + r) * LDSLD + nt * 16 + l16] = acc[nt][r];
}

// ---------------------------------------------------------------------------
// K0a: powers[g][0] = F[g]
__global__ __launch_bounds__(256) void k_power_init(
    const float* __restrict__ F, float* __restrict__ powers) {
  int i = blockIdx.x * blockDim.x + threadIdx.x;
  if (i < G_ * SD * SD) {
    int g = i / (SD * SD), e = i % (SD * SD);
    powers[(size_t)g * LC * SD * SD + e] = F[(size_t)g * SD * SD + e];
  }
}

// K0b: log-doubling level: for j=1..h compute F^{h+j} = F^h * F^j
// grid = (h, G_*4): blockIdx.x = j-1, blockIdx.y = g*4 + mtile
__global__ __launch_bounds__(32) void k_power_level(float* __restrict__ powers,
                                                    int h) {
  const int j  = 1 + blockIdx.x;
  const int g  = blockIdx.y >> 2;
  const int mt = blockIdx.y & 3;
  const float* Gh = powers + ((size_t)g * LC + (h - 1)) * SD * SD;
  const float* Gj = powers + ((size_t)g * LC + (j - 1)) * SD * SD;
  float*       Go = powers + ((size_t)g * LC + (h + j - 1)) * SD * SD;

  v8f acc[4];
#pragma unroll
  for (int nt = 0; nt < 4; ++nt) acc[nt] = vzero8();
#pragma unroll
  for (int kt = 0; kt < 16; ++kt) {
    v2f a = frag_T(Gh, SD, mt * 16, kt);
#pragma unroll
    for (int nt = 0; nt < 4; ++nt)
      acc[nt] = wmma4(a, frag_NT(Gj, SD, nt * 16, kt), acc[nt]);
  }
#pragma unroll
  for (int nt = 0; nt < 4; ++nt)
    ctile_store(Go + mt * 16 * SD, SD, nt * 16, acc[nt]);
}

// ---------------------------------------------------------------------------
// K1: U_t = inputs.B^T + w.S_W^T -> states[t]; V_t = v.S_V^T -> obs[t]
// grid = (L_, G_*8): blockIdx.x = t, blockIdx.y = g*8 + btile
__global__ __launch_bounds__(32) void k_precompute(
    const float* __restrict__ inp, const float* __restrict__ Bm,
    const float* __restrict__ SW, const float* __restrict__ SV,
    const float* __restrict__ w, const float* __restrict__ v,
    float* __restrict__ states, float* __restrict__ obs) {
  const int t  = blockIdx.x;
  const int g  = blockIdx.y >> 3;
  const int b0 = (blockIdx.y & 7) * 16;

  v8f acc[4];
#pragma unroll
  for (int nt = 0; nt < 4; ++nt) acc[nt] = vzero8();

  // w_noise [L,G,B,SD]
  const float* wb = w + ((size_t)t * G_ + g) * B_ * SD + (size_t)b0 * SD;
  const float* SWg = SW + (size_t)g * SD * SD;
#pragma unroll
  for (int kt = 0; kt < 16; ++kt) {
    v2f a = frag_T(wb, SD, 0, kt);
#pragma unroll
    for (int nt = 0; nt < 4; ++nt)
      acc[nt] = wmma4(a, frag_T(SWg, SD, nt * 16, kt), acc[nt]);
  }

  // inputs [G,B,L,ID]: batch-row stride = L_*ID
  const float* ib = inp + ((size_t)(g * B_ + b0)) * L_ * ID + (size_t)t * ID;
  const float* Bg = Bm + (size_t)g * SD * ID;
#pragma unroll
  for (int kt = 0; kt < 4; ++kt) {
    v2f a = frag_T(ib, L_ * ID, 0, kt);
#pragma unroll
    for (int nt = 0; nt < 4; ++nt)
      acc[nt] = wmma4(a, frag_T(Bg, ID, nt * 16, kt), acc[nt]);
  }

  // states [G,B,L,SD]: batch-row stride = L_*SD
  float* sp = states + ((size_t)(g * B_ + b0)) * L_ * SD + (size_t)t * SD;
#pragma unroll
  for (int nt = 0; nt < 4; ++nt) ctile_store(sp, L_ * SD, nt * 16, acc[nt]);

  // colored observation noise -> obs region
  const float* vb = v + ((size_t)t * G_ + g) * B_ * OD + (size_t)b0 * OD;
  const float* SVg = SV + (size_t)g * OD * OD;
  v8f vo = vzero8();
#pragma unroll
  for (int kt = 0; kt < 4; ++kt)
    vo = wmma4(frag_T(vb, OD, 0, kt), frag_T(SVg, OD, 0, kt), vo);
  float* op = obs + ((size_t)(g * B_ + b0)) * L_ * OD + (size_t)t * OD;
  ctile_store(op, L_ * OD, 0, vo);
}

// ---------------------------------------------------------------------------
// K2: per-chunk partial scan with zero initial state (in place over states)
// grid = (NC, G_*8)
__global__ __launch_bounds__(32) void k_chunk_scan(
    const float* __restrict__ F, float* __restrict__ states) {
  const int c  = blockIdx.x;
  const int g  = blockIdx.y >> 3;
  const int b0 = (blockIdx.y & 7) * 16;
  __shared__ float sm[16 * LDSLD];

  // Preload all B fragments of F (x * F^T) into registers: 64 v2f
  const float* Fg = F + (size_t)g * SD * SD;
  v2f bF[16][4];
#pragma unroll
  for (int kt = 0; kt < 16; ++kt)
#pragma unroll
    for (int nt = 0; nt < 4; ++nt) bF[kt][nt] = frag_T(Fg, SD, nt * 16, kt);

  float* sp = states + ((size_t)(g * B_ + b0)) * L_ * SD +
              (size_t)c * LC * SD;
  v8f p[4];
  for (int j = 0; j < LC; ++j) {
    float* tp = sp + (size_t)j * SD;
    v8f acc[4];
#pragma unroll
    for (int nt = 0; nt < 4; ++nt) acc[nt] = ctile_load(tp, L_ * SD, nt * 16);
    if (j > 0) {
      stage4(sm, p);
      __syncthreads();
#pragma unroll
      for (int kt = 0; kt < 16; ++kt) {
        v2f a = frag_T(sm, LDSLD, 0, kt);
#pragma unroll
        for (int nt = 0; nt < 4; ++nt) acc[nt] = wmma4(a, bF[kt][nt], acc[nt]);
      }
      __syncthreads();
    }
#pragma unroll
    for (int nt = 0; nt < 4; ++nt) {
      p[nt] = acc[nt];
      ctile_store(tp, L_ * SD, nt * 16, acc[nt]);
    }
  }
}

// ---------------------------------------------------------------------------
// K3: sequential carries: carry[0]=s0; carry[c]=carry[c-1].(F^LC)^T + p_end
// grid = G_*8 blocks
__global__ __launch_bounds__(32) void k_carry(
    const float* __restrict__ s0, const float* __restrict__ powers,
    const float* __restrict__ states, float* __restrict__ carry) {
  const int g  = blockIdx.x >> 3;
  const int b0 = (blockIdx.x & 7) * 16;
  __shared__ float sm[16 * LDSLD];

  const float* GL = powers + ((size_t)g * LC + (LC - 1)) * SD * SD;  // F^64
  v2f bP[16][4];
#pragma unroll
  for (int kt = 0; kt < 16; ++kt)
#pragma unroll
    for (int nt = 0; nt < 4; ++nt) bP[kt][nt] = frag_T(GL, SD, nt * 16, kt);

  v8f cr[4];
  const float* s0p = s0 + (size_t)(g * B_ + b0) * SD;
#pragma unroll
  for (int nt = 0; nt < 4; ++nt) cr[nt] = ctile_load(s0p, SD, nt * 16);
  float* c0 = carry + ((size_t)0 * G_ + g) * B_ * SD + (size_t)b0 * SD;
#pragma unroll
  for (int nt = 0; nt < 4; ++nt) ctile_store(c0, SD, nt * 16, cr[nt]);

  const float* sbase = states + ((size_t)(g * B_ + b0)) * L_ * SD;
  for (int c = 1; c < NC; ++c) {
    const float* pe = sbase + ((size_t)c * LC - 1) * SD;  // p at chunk c-1 end
    v8f acc[4];
#pragma unroll
    for (int nt = 0; nt < 4; ++nt) acc[nt] = ctile_load(pe, L_ * SD, nt * 16);
    stage4(sm, cr);
    __syncthreads();
#pragma unroll
    for (int kt = 0; kt < 16; ++kt) {
      v2f a = frag_T(sm, LDSLD, 0, kt);
#pragma unroll
      for (int nt = 0; nt < 4; ++nt) acc[nt] = wmma4(a, bP[kt][nt], acc[nt]);
    }
    __syncthreads();
    float* cp = carry + ((size_t)c * G_ + g) * B_ * SD + (size_t)b0 * SD;
#pragma unroll
    for (int nt = 0; nt < 4; ++nt) {
      cr[nt] = acc[nt];
      ctile_store(cp, SD, nt * 16, acc[nt]);
    }
  }
}

// ---------------------------------------------------------------------------
// K4: fixup + projection: s_t = carry[c].(F^{j+1})^T + p_t ; o_t = s_t.H^T + V
// grid = (L_, G_*8)
__global__ __launch_bounds__(32) void k_fixup(
    const float* __restrict__ powers, const float* __restrict__ carry,
    const float* __restrict__ H, float* __restrict__ states,
    float* __restrict__ obs) {
  const int t  = blockIdx.x;
  const int g  = blockIdx.y >> 3;
  const int b0 = (blockIdx.y & 7) * 16;
  const int c  = t / LC;
  const int j  = t % LC;
  __shared__ float sm[16 * LDSLD];

  float* tp = states + ((size_t)(g * B_ + b0)) * L_ * SD + (size_t)t * SD;
  v8f acc[4];
#pragma unroll
  for (int nt = 0; nt < 4; ++nt) acc[nt] = ctile_load(tp, L_ * SD, nt * 16);

  const float* cp = carry + ((size_t)c * G_ + g) * B_ * SD + (size_t)b0 * SD;
  const float* Gp = powers + ((size_t)g * LC + j) * SD * SD;  // F^{j+1}
#pragma unroll
  for (int kt = 0; kt < 16; ++kt) {
    v2f a = frag_T(cp, SD, 0, kt);
#pragma unroll
    for (int nt = 0; nt < 4; ++nt)
      acc[nt] = wmma4(a, frag_T(Gp, SD, nt * 16, kt), acc[nt]);
  }
#pragma unroll
  for (int nt = 0; nt < 4; ++nt) ctile_store(tp, L_ * SD, nt * 16, acc[nt]);

  // observation: o = s.H^T + V (V already in obs region from K1)
  float* op = obs + ((size_t)(g * B_ + b0)) * L_ * OD + (size_t)t * OD;
  v8f vo = ctile_load(op, L_ * OD, 0);
  stage4(sm, acc);
  __syncthreads();
  const float* Hg = H + (size_t)g * OD * SD;
#pragma unroll
  for (int kt = 0; kt < 16; ++kt)
    vo = wmma4(frag_T(sm, LDSLD, 0, kt), frag_T(Hg, SD, 0, kt), vo);
  ctile_store(op, L_ * OD, 0, vo);
}

// ---------------------------------------------------------------------------
extern "C" void kernel_launch(void* const* d_in, const int* in_sizes, int n_in,
                              void* d_out, int out_size, void* d_ws,
                              size_t ws_size, hipStream_t stream) {
  const float* s0  = (const float*)d_in[0];  // [G,B,SD]
  const float* inp = (const float*)d_in[1];  // [G,B,L,ID]
  const float* F   = (const float*)d_in[2];  // [G,SD,SD]
  const float* Bm  = (const float*)d_in[3];  // [G,SD,ID]
  const float* H   = (const float*)d_in[4];  // [G,OD,SD]
  const float* SW  = (const float*)d_in[5];  // [G,SD,SD]
  const float* SV  = (const float*)d_in[6];  // [G,OD,OD]
  const float* w   = (const float*)d_in[7];  // [L,G,B,SD]
  const float* v   = (const float*)d_in[8];  // [L,G,B,OD]

  float* states = (float*)d_out;                              // [G,B,L,SD]
  float* obs    = states + (size_t)G_ * B_ * L_ * SD;         // [G,B,L,OD]

  float* powers = (float*)d_ws;                       // G*LC*SD*SD = 2 MB
  float* carry  = powers + (size_t)G_ * LC * SD * SD; // NC*G*B*SD  = 2 MB
  (void)ws_size; (void)in_sizes; (void)n_in; (void)out_size;

  // K0: F^1..F^64 by log-doubling
  k_power_init<<<(G_ * SD * SD + 255) / 256, 256, 0, stream>>>(F, powers);
  for (int h = 1; h < LC; h <<= 1)
    k_power_level<<<dim3(h, G_ * 4), 32, 0, stream>>>(powers, h);

  // K1: U and colored V (massively parallel, memory bound)
  k_precompute<<<dim3(L_, G_ * 8), 32, 0, stream>>>(inp, Bm, SW, SV, w, v,
                                                    states, obs);
  // K2: per-chunk partial scans (512 waves, depth 64)
  k_chunk_scan<<<dim3(NC, G_ * 8), 32, 0, stream>>>(F, states);
  // K3: carries across chunks (16 waves, depth 32)
  k_carry<<<G_ * 8, 32, 0, stream>>>(s0, powers, states, carry);
  // K4: fixup + projection (massively parallel)
  k_fixup<<<dim3(L_, G_ * 8), 32, 0, stream>>>(powers, carry, H, states, obs);
}